// UnsupLoss_25829933318825
// MI455X (gfx1250) — compile-verified
//
#include <hip/hip_runtime.h>
#include <math.h>

typedef __attribute__((ext_vector_type(2))) float v2f;
typedef __attribute__((ext_vector_type(8))) float v8f;

#define THRES_STATIC 0.2f
#define THRES_DIST   0.1f
#define MARGIN       0.1f

// ws layout: [0]=sum_dyn, [1]=n_dyn, [2]=sum_stat, [3]=n_stat
__global__ void UnsupLoss_init_ws(float* ws) {
    if (threadIdx.x < 4) ws[threadIdx.x] = 0.0f;
}

// One wave = 16 m-columns of one batch. WMMA f32 16x16x4 computes
// D[n,m] = A[n,:]·B[:,m] with A[n] = (x, y, z, ||x_n||^2), B[m] = (-2px, -2py, -2pz, 1),
// so D[n,m] = ||x_n||^2 - 2 x_n·p_m.  min_n dist2 = (min_n D) + ||p_m||^2.
__global__ __launch_bounds__(256) void UnsupLoss_chamfer_kernel(
    const float* __restrict__ x,     // [B,T,N,3]
    const float* __restrict__ y0,    // [B,1,M,3]
    const float* __restrict__ y1,    // [B,1,M,3]
    float* __restrict__ ws,
    int B, int T, int N, int M)
{
    const int lane = threadIdx.x & 31;
    const int wave = (int)((blockIdx.x * blockDim.x + threadIdx.x) >> 5);
    const int tilesPerBatch = M >> 4;
    const int b  = wave / tilesPerBatch;
    const int lm = lane & 15;                     // column within tile
    const int m  = ((wave % tilesPerBatch) << 4) + lm;
    const bool hi = lane >= 16;

    // --- B operand (loop-invariant): column m of the query matrix ---
    const float* p = y0 + ((size_t)b * M + m) * 3;
    const float px = p[0], py = p[1], pz = p[2];
    const float pnorm = px * px + py * py + pz * pz;
    v2f bmat;
    bmat.x = hi ? (-2.0f * pz) : (-2.0f * px);    // K=2 : K=0
    bmat.y = hi ? 1.0f         : (-2.0f * py);    // K=3 : K=1

    // x_t0 = x[:, T-2, :, :3]
    const float* xt0 = x + ((size_t)b * T + (size_t)(T - 2)) * (size_t)N * 3;

    float runmin = 3.4e38f;
    for (int nt = 0; nt < N; nt += 16) {
        const float* q = xt0 + (size_t)(nt + lm) * 3;
        const float qx = q[0], qy = q[1], qz = q[2];
        v2f amat;
        amat.x = hi ? qz : qx;                                     // K=2 : K=0
        amat.y = hi ? (qx * qx + qy * qy + qz * qz) : qy;          // K=3 : K=1

        v8f acc = {};
        acc = __builtin_amdgcn_wmma_f32_16x16x4_f32(
            /*neg_a=*/false, amat, /*neg_b=*/false, bmat,
            /*c_mod=*/(short)0, acc, /*reuse_a=*/false, /*reuse_b=*/false);

        // per-lane min over the 8 rows this lane holds
        float t0 = fminf(acc[0], acc[1]);
        float t1 = fminf(acc[2], acc[3]);
        float t2 = fminf(acc[4], acc[5]);
        float t3 = fminf(acc[6], acc[7]);
        float m8 = fminf(fminf(t0, t1), fminf(t2, t3));
        runmin = fminf(runmin, m8);
    }
    // combine the two half-wave row groups (lanes l and l^16 share a column)
    runmin = fminf(runmin, __shfl_xor(runmin, 16, 32));
    const float dist2 = runmin + pnorm;

    // --- per-point loss contributions (lanes 0-15 only; 16-31 are duplicates) ---
    float s_dyn = 0.0f, c_dyn = 0.0f, s_stat = 0.0f, c_stat = 0.0f;
    if (!hi) {
        const float* a1 = y1 + ((size_t)b * M + m) * 3;
        const float dx = a1[0] - px;
        const float dy = a1[1] - py;
        const float dz = a1[2] - pz;
        if (dist2 < THRES_DIST) {
            const float sq = dx * dx + dy * dy + dz * dz;
            if (sq > 0.0f) {
                s_dyn = log1pf(expf(MARGIN - sqrtf(sq)));  // softplus(0.1 - norm)
                c_dyn = 1.0f;
            }
        }
        if (dist2 > THRES_STATIC) {
            if (dx > 0.0f) { s_stat += dx * dx; c_stat += 1.0f; }
            if (dy > 0.0f) { s_stat += dy * dy; c_stat += 1.0f; }
            if (dz > 0.0f) { s_stat += dz * dz; c_stat += 1.0f; }
        }
    }

    // 32-lane butterfly reduction
    #pragma unroll
    for (int off = 16; off > 0; off >>= 1) {
        s_dyn  += __shfl_xor(s_dyn,  off, 32);
        c_dyn  += __shfl_xor(c_dyn,  off, 32);
        s_stat += __shfl_xor(s_stat, off, 32);
        c_stat += __shfl_xor(c_stat, off, 32);
    }
    if (lane == 0) {
        __hip_atomic_fetch_add(&ws[0], s_dyn,  __ATOMIC_RELAXED, __HIP_MEMORY_SCOPE_AGENT);
        __hip_atomic_fetch_add(&ws[1], c_dyn,  __ATOMIC_RELAXED, __HIP_MEMORY_SCOPE_AGENT);
        __hip_atomic_fetch_add(&ws[2], s_stat, __ATOMIC_RELAXED, __HIP_MEMORY_SCOPE_AGENT);
        __hip_atomic_fetch_add(&ws[3], c_stat, __ATOMIC_RELAXED, __HIP_MEMORY_SCOPE_AGENT);
    }
}

__global__ void UnsupLoss_finalize(const float* __restrict__ ws, float* __restrict__ out) {
    if (threadIdx.x == 0) {
        out[0] = ws[0] / fmaxf(ws[1], 1.0f);  // loss_dynamic
        out[1] = ws[2] / fmaxf(ws[3], 1.0f);  // loss_static
    }
}

extern "C" void kernel_launch(void* const* d_in, const int* in_sizes, int n_in,
                              void* d_out, int out_size, void* d_ws, size_t ws_size,
                              hipStream_t stream) {
    const float* x  = (const float*)d_in[0];   // [8,4,4096,3]
    const float* y0 = (const float*)d_in[1];   // [8,1,4096,3]
    const float* y1 = (const float*)d_in[2];   // [8,1,4096,3]
    float* out = (float*)d_out;
    float* ws  = (float*)d_ws;

    const int B = 8, T = 4, N = 4096;
    const int M = in_sizes[1] / (B * 3);       // 4096

    UnsupLoss_init_ws<<<1, 32, 0, stream>>>(ws);

    const int waves   = B * (M / 16);          // 2048 waves
    const int threads = 256;                   // 8 waves per block
    const int blocks  = (waves * 32) / threads;
    UnsupLoss_chamfer_kernel<<<blocks, threads, 0, stream>>>(x, y0, y1, ws, B, T, N, M);

    UnsupLoss_finalize<<<1, 32, 0, stream>>>(ws, out);
}